// RelationHead_24300924961313
// MI455X (gfx1250) — compile-verified
//
#include <hip/hip_runtime.h>
#include <math.h>

#define N_INST 64
#define NPIX   64000      // 200*320

typedef __attribute__((ext_vector_type(2))) float v2f;
typedef __attribute__((ext_vector_type(8))) float v8f;

// Fast sigmoid: v_mul (log2e scale inside __expf) + v_exp_f32 + v_add + v_rcp_f32.
// Avoids the correctly-rounded IEEE divide expansion (div_scale/div_fmas/div_fixup).
__device__ __forceinline__ float sigmoidf(float x) {
    return __builtin_amdgcn_rcpf(1.0f + __expf(-x));
}

// ---------------------------------------------------------------------------
// Kernel A: class features.  W1[c][i] = relu(U_w[c, a_i]) * P_w[c],
//           F2[c][i] = relu(V_w[c, a_i]),  a_i = cls_idx[i]-1.
// Stored column-major (c-major, 64 lanes of i) so the consumer reads coalesced.
// ---------------------------------------------------------------------------
__global__ __launch_bounds__(256) void feat_kernel(
    const int* __restrict__ cls, const float* __restrict__ U_w,
    const float* __restrict__ V_w, const float* __restrict__ P_w,
    float* __restrict__ W1, float* __restrict__ F2) {
    int gid = blockIdx.x * 256 + threadIdx.x;   // 0 .. 16383
    int c = gid >> 6;
    int i = gid & 63;
    int a = cls[i] - 1;
    float u = U_w[c * 80 + a];
    float v = V_w[c * 80 + a];
    W1[c * 64 + i] = fmaxf(u, 0.0f) * P_w[c];
    F2[c * 64 + i] = fmaxf(v, 0.0f);
}

// ---------------------------------------------------------------------------
// Kernel B: R[i][j] = sigmoid( sum_c W1[c][i]*F2[c][j]
//                            + sum_o relu(<bbox_mat[i,j], Wc[o]>) * P[256+o] )
// One thread per (i,j).  i = lane-varying -> W1 reads coalesced, F2/j uniform
// per wave -> scalar-broadcast loads.
// ---------------------------------------------------------------------------
__global__ __launch_bounds__(256) void logit_kernel(
    const float* __restrict__ W1, const float* __restrict__ F2,
    const float* __restrict__ bbox, const float* __restrict__ Wc_w,
    const float* __restrict__ P_w, float* __restrict__ R) {
    int gid = blockIdx.x * 256 + threadIdx.x;   // 0 .. 4095
    int i = gid & 63;
    int j = gid >> 6;

    float acc = 0.0f;
    for (int c = 0; c < 256; ++c)
        acc += W1[c * 64 + i] * F2[c * 64 + j];

    float xi0 = bbox[i * 5 + 1], yi0 = bbox[i * 5 + 2];
    float xi1 = bbox[i * 5 + 3], yi1 = bbox[i * 5 + 4];
    float xj0 = bbox[j * 5 + 1], yj0 = bbox[j * 5 + 2];
    float xj1 = bbox[j * 5 + 3], yj1 = bbox[j * 5 + 4];
    float wi = xi1 - xi0 + 1.0f, hi = yi1 - yi0 + 1.0f;
    float wj = xj1 - xj0 + 1.0f, hj = yj1 - yj0 + 1.0f;
    float xci = xi0 + 0.5f * wi, yci = yi0 + 0.5f * hi;
    float xcj = xj0 + 0.5f * wj, ycj = yj0 + 0.5f * hj;
    float rwi = __builtin_amdgcn_rcpf(wi);
    float rhi = __builtin_amdgcn_rcpf(hi);
    float dx = -(xci - xcj) * rwi;
    float dy = -(yci - ycj) * rhi;
    float dw = __logf(wj * rwi);
    float dh = __logf(hj * rhi);

    for (int o = 0; o < 128; ++o) {
        float v = dx * Wc_w[o * 4 + 0] + dy * Wc_w[o * 4 + 1] +
                  dw * Wc_w[o * 4 + 2] + dh * Wc_w[o * 4 + 3];
        acc += fmaxf(v, 0.0f) * P_w[256 + o];
    }
    R[i * 64 + j] = sigmoidf(acc);
}

// ---------------------------------------------------------------------------
// Kernel C: O[i][j] = relu(R[i][j] - R[j][i]); write to workspace (for the
// GEMM) and to the second output slot.
// ---------------------------------------------------------------------------
__global__ __launch_bounds__(256) void omat_kernel(
    const float* __restrict__ R, float* __restrict__ O_ws,
    float* __restrict__ O_out) {
    int gid = blockIdx.x * 256 + threadIdx.x;   // gid == i*64 + j
    int i = gid >> 6;
    int j = gid & 63;
    float o = fmaxf(R[i * 64 + j] - R[j * 64 + i], 0.0f);
    O_ws[gid]  = o;
    O_out[gid] = o;
}

// ---------------------------------------------------------------------------
// Kernel D: the heavy pass.
//   D[i,p] = sum_j O[i,j] * sigmoid(mask[j,p])        (fp32 WMMA 16x16x4)
//   out[i,p] = mask[i,p] * (1 - sigmoid(mask[i,p]) * D[i,p])
// Block = 256 threads = 8 waves; each wave owns 16 pixel columns and all 64
// output rows (4 WMMA accumulator tiles).  Grid = 64000/128 = 500 blocks.
// A/B lane layout per CDNA5 ISA: lanes 0-15 carry K=k0+{0,1}, lanes 16-31
// carry K=k0+{2,3}; C/D VGPR r -> rows r / r+8.
// ---------------------------------------------------------------------------
__global__ __launch_bounds__(256) void apply_kernel(
    const float* __restrict__ mask, const float* __restrict__ O,
    float* __restrict__ out) {
    __shared__ float sO[64][68];   // stride 68: conflict-free A reads

    for (int idx = threadIdx.x; idx < 64 * 64; idx += 256)
        sO[idx >> 6][idx & 63] = O[idx];
    __syncthreads();

    const int lane = threadIdx.x & 31;
    const int wave = threadIdx.x >> 5;
    const int col  = lane & 15;    // N index within the 16-wide tile
    const int half = lane >> 4;    // 0: K=k0+{0,1}   1: K=k0+{2,3}
    const int p    = blockIdx.x * 128 + wave * 16 + col;

    v8f acc[4] = {v8f{}, v8f{}, v8f{}, v8f{}};

    #pragma unroll
    for (int k0 = 0; k0 < 64; k0 += 4) {
        const int kA = k0 + half * 2;
        v2f b;
        b[0] = sigmoidf(mask[(kA + 0) * NPIX + p]);
        b[1] = sigmoidf(mask[(kA + 1) * NPIX + p]);
        #pragma unroll
        for (int mt = 0; mt < 4; ++mt) {
            v2f a;
            a[0] = sO[mt * 16 + col][kA + 0];
            a[1] = sO[mt * 16 + col][kA + 1];
            acc[mt] = __builtin_amdgcn_wmma_f32_16x16x4_f32(
                false, a, false, b, (short)0, acc[mt], false, false);
        }
    }

    // Epilogue: new_mask = m * (1 - sigmoid(m) * weighted)
    #pragma unroll
    for (int mt = 0; mt < 4; ++mt) {
        #pragma unroll
        for (int r = 0; r < 8; ++r) {
            int row = mt * 16 + half * 8 + r;
            float m  = mask[row * NPIX + p];
            float tm = sigmoidf(m);
            out[row * NPIX + p] = m * (1.0f - tm * acc[mt][r]);
        }
    }
}

// ---------------------------------------------------------------------------
extern "C" void kernel_launch(void* const* d_in, const int* in_sizes, int n_in,
                              void* d_out, int out_size, void* d_ws, size_t ws_size,
                              hipStream_t stream) {
    (void)in_sizes; (void)n_in; (void)out_size; (void)ws_size;

    const float* mask = (const float*)d_in[0];   // (1,64,200,320)
    const float* bbox = (const float*)d_in[1];   // (64,5)
    const int*   cls  = (const int*)  d_in[2];   // (64,)
    const float* U_w  = (const float*)d_in[3];   // (256,80)
    const float* V_w  = (const float*)d_in[4];   // (256,80)
    const float* Wc_w = (const float*)d_in[5];   // (128,4)
    const float* P_w  = (const float*)d_in[6];   // (1,384)

    float* out = (float*)d_out;                  // new_mask (4096000) + O (4096)
    float* ws  = (float*)d_ws;
    float* W1  = ws;                             // 16384 floats
    float* F2  = ws + 16384;                     // 16384 floats
    float* R   = ws + 32768;                     // 4096 floats
    float* Omat= ws + 36864;                     // 4096 floats

    feat_kernel <<<64,  256, 0, stream>>>(cls, U_w, V_w, P_w, W1, F2);
    logit_kernel<<<16,  256, 0, stream>>>(W1, F2, bbox, Wc_w, P_w, R);
    omat_kernel <<<16,  256, 0, stream>>>(R, Omat, out + (size_t)N_INST * NPIX);
    apply_kernel<<<500, 256, 0, stream>>>(mask, Omat, out);
}